// EventPointNet_28260884807847
// MI455X (gfx1250) — compile-verified
//
#include <hip/hip_runtime.h>

typedef float v2f __attribute__((ext_vector_type(2)));
typedef float v8f __attribute__((ext_vector_type(8)));

#define NNODES 131072
#define NEDGES 2097152
#define BN_EPS 1e-5f

// ---------------------------------------------------------------------------
// Dense GEMM: Y[n,COUT] = X[n,CIN] @ Wp[CIN,WP], via V_WMMA_F32_16X16X4_F32.
// Wp is the weight matrix zero-padded to WP = ceil(COUT/16)*16 columns, and
// (for CIN==1) K-padding rows are zero too, so all B loads are unconditional.
// One wave computes a 16-row tile across all WP columns; fully unrolled so
// accumulators are static VGPRs (no movrel) and loads are b64.
// ---------------------------------------------------------------------------
template <int CIN, int COUT>
__global__ __launch_bounds__(256) void k_gemm_wmma(const float* __restrict__ X,
                                                   const float* __restrict__ Wp,
                                                   float* __restrict__ Y, int n) {
  constexpr int NT = (COUT + 15) / 16;
  constexpr int WP = NT * 16;
  int lane = threadIdx.x & 31;
  int wave = threadIdx.x >> 5;
  int rowBase = (blockIdx.x * 8 + wave) * 16;   // uniform per wave
  if (rowBase >= n) return;                      // wave-uniform: EXEC stays all-ones

  int half = lane >> 4;   // 0: K=kb..kb+1, 1: K=kb+2..kb+3
  int lid  = lane & 15;
  int row  = rowBase + lid;

  v8f acc[NT];
#pragma unroll
  for (int t = 0; t < NT; ++t) acc[t] = (v8f)(0.0f);

  if constexpr (CIN == 1) {
    float xv = X[row];                 // unconditional, in-bounds
    v2f a;
    a.x = half ? 0.0f : xv;            // K=2 (half=1) and K=1 don't exist -> 0
    a.y = 0.0f;
#pragma unroll
    for (int t = 0; t < NT; ++t) {
      int col = t * 16 + lid;
      // Wp padded rows are zero, so K=1..3 B entries read zeros: harmless.
      const v2f* pw = (const v2f*)(Wp + (size_t)(2 * half) * WP + col - lid);
      v2f b;
      b.x = Wp[(size_t)(2 * half) * WP + col];
      b.y = Wp[(size_t)(2 * half + 1) * WP + col];
      (void)pw;
      acc[t] = __builtin_amdgcn_wmma_f32_16x16x4_f32(
          false, a, false, b, (short)0, acc[t], false, false);
    }
  } else {
    static_assert(CIN % 4 == 0, "CIN must be a multiple of 4 here");
    const v2f* px = (const v2f*)(X + (size_t)row * CIN);   // 8B-aligned pairs
#pragma unroll
    for (int kb = 0; kb < CIN; kb += 4) {
      int k0 = kb + 2 * half;
      v2f a = px[k0 >> 1];
#pragma unroll
      for (int t = 0; t < NT; ++t) {
        int col = t * 16 + lid;
        v2f b;
        b.x = Wp[(size_t)k0 * WP + col];
        b.y = Wp[(size_t)(k0 + 1) * WP + col];
        acc[t] = __builtin_amdgcn_wmma_f32_16x16x4_f32(
            false, a, false, b, (short)0, acc[t], false, false);
      }
    }
  }

#pragma unroll
  for (int t = 0; t < NT; ++t) {
    int col = t * 16 + lid;
    if (COUT == WP || col < COUT) {     // compile-time true except COUT=8 and COUT=17 tail tiles
#pragma unroll
      for (int r = 0; r < 8; ++r) {
        int m = r + 8 * half;
        Y[(size_t)(rowBase + m) * COUT + col] = acc[t][r];
      }
    }
  }
}

// Zero-fill + copy W[cin,cout] into padded Wp[64,wp] (wp = ceil(cout/16)*16).
__global__ void k_padW(const float* __restrict__ W, float* __restrict__ Wp,
                       int cin, int cout, int wp) {
  int i = blockIdx.x * blockDim.x + threadIdx.x;
  if (i < 4096) Wp[i] = 0.0f;
  if (i < cin * cout) {
    int k = i / cout, j = i - k * cout;
    // note: separate pass ordering hazard avoided because wp*k+j < 4096 and the
    // zero above covers a superset; same-thread write-after-write is fine only
    // within a thread, so do the copy in a second kernel instead.
  }
}

__global__ void k_copyW(const float* __restrict__ W, float* __restrict__ Wp,
                        int cin, int cout, int wp) {
  int i = blockIdx.x * blockDim.x + threadIdx.x;
  if (i < cin * cout) {
    int k = i / cout, j = i - k * cout;
    Wp[(size_t)k * wp + j] = W[i];
  }
}

// ---------------------------------------------------------------------------
// Small utility kernels
// ---------------------------------------------------------------------------
__global__ void k_copy_edges(const int* __restrict__ eidx, int* src, int* dst,
                             float* em, int E) {
  int i = blockIdx.x * blockDim.x + threadIdx.x;
  if (i < E) { src[i] = eidx[i]; dst[i] = eidx[E + i]; em[i] = 1.0f; }
}

__global__ void k_zero_f(float* p, int n) {
  int i = blockIdx.x * blockDim.x + threadIdx.x;
  if (i < n) p[i] = 0.0f;
}

__global__ void k_deg_init(float* deg, int n) {
  int i = blockIdx.x * blockDim.x + threadIdx.x;
  if (i < n) deg[i] = 2.0f;   // improved=True self-loop weight
}

__global__ void k_deg_acc(const int* __restrict__ dst, const float* __restrict__ em,
                          float* deg, int E) {
  int i = blockIdx.x * blockDim.x + threadIdx.x;
  if (i < E) atomicAdd(&deg[dst[i]], em[i]);
}

__global__ void k_dinv(const float* deg, float* dinv, int n) {
  int i = blockIdx.x * blockDim.x + threadIdx.x;
  if (i < n) dinv[i] = rsqrtf(deg[i]);
}

__global__ void k_self_init(const float* __restrict__ xw, const float* __restrict__ dinv,
                            const float* __restrict__ bias, float* y, int n, int cout) {
  int i = blockIdx.x * blockDim.x + threadIdx.x;
  if (i < n * cout) {
    int node = i / cout, c = i - node * cout;
    float dv = dinv[node];
    y[i] = bias[c] + 2.0f * dv * dv * xw[i];
  }
}

__global__ void k_edge_agg(const int* __restrict__ src, const int* __restrict__ dst,
                           const float* __restrict__ em, const float* __restrict__ dinv,
                           const float* __restrict__ xw, float* y, int E, int cout) {
  int e = blockIdx.x * blockDim.x + threadIdx.x;
  if (e < E) {
    float m = em[e];
    if (m != 0.0f) {
      int s = src[e], d = dst[e];
      float coef = dinv[s] * dinv[d] * m;
      const float* xs = xw + (size_t)s * cout;
      float* yd = y + (size_t)d * cout;
      __builtin_prefetch(xs, 0, 0);     // global_prefetch_b8 on the gathered row
      for (int c = 0; c < cout; ++c) atomicAdd(&yd[c], coef * xs[c]);
    }
  }
}

// ELU in place + per-channel sum / sum-of-squares (LDS partials -> global atomics)
__global__ __launch_bounds__(256) void k_elu_stats(float* y, float* sum, float* sumsq,
                                                   int total, int cout) {
  __shared__ float ssum[64];
  __shared__ float ssq[64];
  if (threadIdx.x < (unsigned)cout) { ssum[threadIdx.x] = 0.0f; ssq[threadIdx.x] = 0.0f; }
  __syncthreads();
  int stride = gridDim.x * blockDim.x;
  for (int i = blockIdx.x * blockDim.x + threadIdx.x; i < total; i += stride) {
    float v = y[i];
    v = v > 0.0f ? v : expm1f(v);       // jax.nn.elu, alpha=1
    y[i] = v;
    int c = i % cout;
    atomicAdd(&ssum[c], v);
    atomicAdd(&ssq[c], v * v);
  }
  __syncthreads();
  if (threadIdx.x < (unsigned)cout) {
    atomicAdd(&sum[threadIdx.x], ssum[threadIdx.x]);
    atomicAdd(&sumsq[threadIdx.x], ssq[threadIdx.x]);
  }
}

__global__ void k_bn_finalize(const float* sum, const float* sumsq, const float* g,
                              const float* b, float* ss, int n, int cout) {
  int c = threadIdx.x;
  if (c < cout) {
    float inv_n = 1.0f / (float)n;
    float m = sum[c] * inv_n;
    float v = sumsq[c] * inv_n - m * m;
    float sc = g[c] * rsqrtf(v + BN_EPS);
    ss[c] = sc;
    ss[cout + c] = b[c] - m * sc;
  }
}

__global__ void k_bn_apply(const float* __restrict__ y, const float* __restrict__ ss,
                           float* out, int total, int cout) {
  int i = blockIdx.x * blockDim.x + threadIdx.x;
  if (i < total) {
    int c = i % cout;
    out[i] = y[i] * ss[c] + ss[cout + c];
  }
}

// ---------------------------------------------------------------------------
// TopK pooling: scores, deterministic radix-select, scans, compaction, remap
// ---------------------------------------------------------------------------
__global__ void k_wnorm(const float* w, float* invnorm, int cout) {
  __shared__ float s[64];
  int t = threadIdx.x;
  s[t] = (t < cout) ? w[t] * w[t] : 0.0f;
  __syncthreads();
  for (int o = 32; o > 0; o >>= 1) { if (t < o) s[t] += s[t + o]; __syncthreads(); }
  if (t == 0) invnorm[0] = rsqrtf(s[0]);
}

__global__ void k_score(const float* __restrict__ x, const float* __restrict__ w,
                        const float* __restrict__ invnorm, float* sc, unsigned* keys,
                        int n, int cout) {
  int i = blockIdx.x * blockDim.x + threadIdx.x;
  if (i < n) {
    const float* xr = x + (size_t)i * cout;
    float d = 0.0f;
    for (int c = 0; c < cout; ++c) d += xr[c] * w[c];
    float s = tanhf(d * invnorm[0]);
    sc[i] = s;
    unsigned u = __float_as_uint(s);
    u = (u & 0x80000000u) ? ~u : (u | 0x80000000u);   // monotone map
    keys[i] = ~u;                                      // ascending key == descending score
  }
}

// state[0]=remaining rank, state[1]=prefix/threshold, state[2]=count_less, state[3]=need
__global__ void k_sel_init(int* state, unsigned* hist, int k) {
  int t = threadIdx.x;
  if (t < 256) hist[t] = 0u;
  if (t == 0) { state[0] = k; state[1] = 0; state[2] = 0; state[3] = 0; }
}

__global__ void k_hist(const unsigned* __restrict__ keys, unsigned* hist,
                       const int* __restrict__ state, int n, int pass) {
  int i = blockIdx.x * blockDim.x + threadIdx.x;
  if (i < n) {
    unsigned u = keys[i];
    unsigned pfx = (unsigned)state[1];
    int hs = (pass + 1) * 8;
    bool match = (hs >= 32) || ((u >> hs) == (pfx >> hs));
    if (match) atomicAdd(&hist[(u >> (pass * 8)) & 255u], 1u);
  }
}

__global__ void k_select(unsigned* hist, int* state, int pass) {
  if (threadIdx.x == 0) {
    unsigned rem = (unsigned)state[0];
    unsigned cum = 0;
    int dig = 255;
    for (int d = 0; d < 256; ++d) {
      unsigned c = hist[d];
      if (cum + c >= rem) { dig = d; break; }
      cum += c;
    }
    state[0] = (int)(rem - cum);
    state[1] |= dig << (pass * 8);
  }
  __syncthreads();
  if (threadIdx.x < 256) hist[threadIdx.x] = 0u;   // ready for next pass
}

__global__ void k_count_less(const unsigned* keys, int* state, int n) {
  int i = blockIdx.x * blockDim.x + threadIdx.x;
  if (i < n && keys[i] < (unsigned)state[1]) atomicAdd(&state[2], 1);
}

__global__ void k_need(int* state, int k) {
  if (threadIdx.x == 0 && blockIdx.x == 0) state[3] = k - state[2];
}

__global__ void k_eqflag(const unsigned* keys, const int* state, int* eq, int n) {
  int i = blockIdx.x * blockDim.x + threadIdx.x;
  if (i < n) eq[i] = (keys[i] == (unsigned)state[1]) ? 1 : 0;
}

__global__ __launch_bounds__(256) void k_scan_partial(const int* __restrict__ in,
                                                      int* outExcl, int* bsum, int n) {
  __shared__ int s[256];
  int i = blockIdx.x * 256 + threadIdx.x;
  int v = (i < n) ? in[i] : 0;
  s[threadIdx.x] = v;
  __syncthreads();
  for (int o = 1; o < 256; o <<= 1) {
    int t = (threadIdx.x >= (unsigned)o) ? s[threadIdx.x - o] : 0;
    __syncthreads();
    s[threadIdx.x] += t;
    __syncthreads();
  }
  if (i < n) outExcl[i] = s[threadIdx.x] - v;
  if (threadIdx.x == 255) bsum[blockIdx.x] = s[255];
}

__global__ __launch_bounds__(1024) void k_scan_bsums(int* bsum, int nb) {
  __shared__ int s[1024];
  int t = threadIdx.x;
  int v = (t < nb) ? bsum[t] : 0;
  s[t] = v;
  __syncthreads();
  for (int o = 1; o < 1024; o <<= 1) {
    int tv = (t >= o) ? s[t - o] : 0;
    __syncthreads();
    s[t] += tv;
    __syncthreads();
  }
  if (t < nb) bsum[t] = s[t] - v;   // exclusive
}

__global__ void k_scan_add(int* outExcl, const int* bsum, int n) {
  int i = blockIdx.x * 256 + threadIdx.x;
  if (i < n) outExcl[i] += bsum[blockIdx.x];
}

__global__ void k_keep(const unsigned* keys, const int* eqrank, const int* state,
                       int* keep, int n) {
  int i = blockIdx.x * blockDim.x + threadIdx.x;
  if (i < n) {
    unsigned T = (unsigned)state[1];
    unsigned u = keys[i];
    keep[i] = (u < T) || (u == T && eqrank[i] < state[3]);
  }
}

__global__ void k_compact(const float* __restrict__ x, const float* __restrict__ sc,
                          const int* __restrict__ keep, const int* __restrict__ pos,
                          float* xn, int* newid, int n, int cout) {
  int i = blockIdx.x * blockDim.x + threadIdx.x;
  if (i < n) {
    if (keep[i]) {
      int p = pos[i];
      newid[i] = p;
      float s = sc[i];
      for (int c = 0; c < cout; ++c)
        xn[(size_t)p * cout + c] = x[(size_t)i * cout + c] * s;
    } else {
      newid[i] = -1;
    }
  }
}

__global__ void k_remap(int* src, int* dst, float* em, const int* __restrict__ newid, int E) {
  int e = blockIdx.x * blockDim.x + threadIdx.x;
  if (e < E) {
    int s2 = newid[src[e]];
    int d2 = newid[dst[e]];
    if (s2 < 0 || d2 < 0) em[e] = 0.0f;
    src[e] = s2 >= 0 ? s2 : 0;
    dst[e] = d2 >= 0 ? d2 : 0;
  }
}

// ---------------------------------------------------------------------------
// Host orchestration
// ---------------------------------------------------------------------------
static inline int cdiv(int a, int b) { return (a + b - 1) / b; }

static void launch_gemm(const float* X, const float* Wp, float* Y, int n,
                        int cin, int cout, hipStream_t s) {
  int grid = cdiv(n, 128);
  if      (cin == 1  && cout == 8)  k_gemm_wmma<1, 8><<<grid, 256, 0, s>>>(X, Wp, Y, n);
  else if (cin == 8  && cout == 8)  k_gemm_wmma<8, 8><<<grid, 256, 0, s>>>(X, Wp, Y, n);
  else if (cin == 8  && cout == 16) k_gemm_wmma<8, 16><<<grid, 256, 0, s>>>(X, Wp, Y, n);
  else if (cin == 16 && cout == 16) k_gemm_wmma<16, 16><<<grid, 256, 0, s>>>(X, Wp, Y, n);
  else if (cin == 16 && cout == 32) k_gemm_wmma<16, 32><<<grid, 256, 0, s>>>(X, Wp, Y, n);
  else if (cin == 32 && cout == 32) k_gemm_wmma<32, 32><<<grid, 256, 0, s>>>(X, Wp, Y, n);
  else if (cin == 32 && cout == 64) k_gemm_wmma<32, 64><<<grid, 256, 0, s>>>(X, Wp, Y, n);
  else if (cin == 64 && cout == 64) k_gemm_wmma<64, 64><<<grid, 256, 0, s>>>(X, Wp, Y, n);
  else if (cin == 64 && cout == 17) k_gemm_wmma<64, 17><<<grid, 256, 0, s>>>(X, Wp, Y, n);
}

static void scan_excl(const int* in, int* out, int* bsum, int n, hipStream_t stream) {
  int nb = cdiv(n, 256);
  k_scan_partial<<<nb, 256, 0, stream>>>(in, out, bsum, n);
  k_scan_bsums<<<1, 1024, 0, stream>>>(bsum, nb);
  k_scan_add<<<nb, 256, 0, stream>>>(out, bsum, n);
}

static void run_conv(const float* x, float* y, float* dest, float* xw, float* Wpad,
                     float* deg, float* dinv, const int* srcW, const int* dstW,
                     const float* emW, float* accstats, float* ssbuf,
                     const float* W, const float* b, const float* g, const float* bb,
                     int n, int cin, int cout, hipStream_t stream) {
  int total = n * cout;
  int blocksE = cdiv(NEDGES, 256);
  int wp = cdiv(cout, 16) * 16;
  k_zero_f<<<16, 256, 0, stream>>>(Wpad, 4096);
  k_copyW<<<cdiv(cin * cout, 256), 256, 0, stream>>>(W, Wpad, cin, cout, wp);
  k_zero_f<<<1, 256, 0, stream>>>(accstats, 128);
  k_deg_init<<<cdiv(n, 256), 256, 0, stream>>>(deg, n);
  launch_gemm(x, Wpad, xw, n, cin, cout, stream);
  k_deg_acc<<<blocksE, 256, 0, stream>>>(dstW, emW, deg, NEDGES);
  k_dinv<<<cdiv(n, 256), 256, 0, stream>>>(deg, dinv, n);
  k_self_init<<<cdiv(total, 256), 256, 0, stream>>>(xw, dinv, b, y, n, cout);
  k_edge_agg<<<blocksE, 256, 0, stream>>>(srcW, dstW, emW, dinv, xw, y, NEDGES, cout);
  int sgrid = cdiv(total, 256); if (sgrid > 2048) sgrid = 2048;
  k_elu_stats<<<sgrid, 256, 0, stream>>>(y, accstats, accstats + 64, total, cout);
  k_bn_finalize<<<1, 64, 0, stream>>>(accstats, accstats + 64, g, bb, ssbuf, n, cout);
  k_bn_apply<<<cdiv(total, 256), 256, 0, stream>>>(y, ssbuf, dest, total, cout);
}

extern "C" void kernel_launch(void* const* d_in, const int* in_sizes, int n_in,
                              void* d_out, int out_size, void* d_ws, size_t ws_size,
                              hipStream_t stream) {
  (void)in_sizes; (void)n_in; (void)out_size; (void)ws_size;
  static const int CIN[9]  = {1, 8, 8, 16, 16, 32, 32, 64, 64};
  static const int COUT[9] = {8, 8, 16, 16, 32, 32, 64, 64, 17};

  const float* x0  = (const float*)d_in[0];
  const int*   eix = (const int*)d_in[1];
  const float *convW[9], *convB[9], *normG[9], *normB[9], *poolW[4];
  int idx = 2;
  for (int i = 0; i < 9; ++i) convW[i] = (const float*)d_in[idx++];
  for (int i = 0; i < 9; ++i) convB[i] = (const float*)d_in[idx++];
  for (int i = 0; i < 9; ++i) normG[i] = (const float*)d_in[idx++];
  for (int i = 0; i < 9; ++i) normB[i] = (const float*)d_in[idx++];
  for (int i = 0; i < 4; ++i) poolW[i] = (const float*)d_in[idx++];

  char* ws = (char*)d_ws;
  size_t off = 0;
  auto alloc = [&](size_t bytes) -> void* {
    void* p = ws + off;
    off += (bytes + 255) & ~(size_t)255;
    return p;
  };
  const size_t XMAX = 1048576;   // max n*cout over all stages (exactly 2^20)
  float*    xa       = (float*)alloc(XMAX * 4);
  float*    xb       = (float*)alloc(XMAX * 4);
  float*    xw       = (float*)alloc(XMAX * 4);
  float*    Wpad     = (float*)alloc(4096 * 4);   // 64x64 zero-padded weights
  float*    deg      = (float*)alloc(NNODES * 4);
  float*    dinv     = (float*)alloc(NNODES * 4);
  float*    sc       = (float*)alloc(NNODES * 4);
  unsigned* keys     = (unsigned*)alloc(NNODES * 4);
  int*      eq       = (int*)alloc(NNODES * 4);
  int*      eqrank   = (int*)alloc(NNODES * 4);
  int*      keepf    = (int*)alloc(NNODES * 4);
  int*      pos      = (int*)alloc(NNODES * 4);
  int*      newid    = (int*)alloc(NNODES * 4);
  int*      bsum     = (int*)alloc(1024 * 4);
  unsigned* hist     = (unsigned*)alloc(256 * 4);
  int*      state    = (int*)alloc(16 * 4);
  float*    invnorm  = (float*)alloc(256);
  float*    accstats = (float*)alloc(128 * 4);   // [0..63]=sum, [64..127]=sumsq
  float*    ssbuf    = (float*)alloc(128 * 4);   // [0..cout)=scale, [cout..2cout)=shift
  int*      srcW     = (int*)alloc((size_t)NEDGES * 4);
  int*      dstW     = (int*)alloc((size_t)NEDGES * 4);
  float*    emW      = (float*)alloc((size_t)NEDGES * 4);

  k_copy_edges<<<cdiv(NEDGES, 256), 256, 0, stream>>>(eix, srcW, dstW, emW, NEDGES);

  float* bufs[2] = {xa, xb};
  const float* xcur = x0;
  int cur = 0;
  int n = NNODES;
  int ci = 0;

  for (int s = 0; s < 4; ++s) {
    for (int r = 0; r < 2; ++r) {
      float* y = bufs[cur];
      run_conv(xcur, y, y, xw, Wpad, deg, dinv, srcW, dstW, emW, accstats, ssbuf,
               convW[ci], convB[ci], normG[ci], normB[ci], n, CIN[ci], COUT[ci], stream);
      xcur = y; cur ^= 1; ++ci;
    }
    int cout = COUT[ci - 1];
    int k = n / 2;
    // TopK pooling
    k_wnorm<<<1, 64, 0, stream>>>(poolW[s], invnorm, cout);
    k_score<<<cdiv(n, 256), 256, 0, stream>>>(xcur, poolW[s], invnorm, sc, keys, n, cout);
    k_sel_init<<<1, 256, 0, stream>>>(state, hist, k);
    for (int pass = 3; pass >= 0; --pass) {
      k_hist<<<cdiv(n, 256), 256, 0, stream>>>(keys, hist, state, n, pass);
      k_select<<<1, 256, 0, stream>>>(hist, state, pass);
    }
    k_count_less<<<cdiv(n, 256), 256, 0, stream>>>(keys, state, n);
    k_need<<<1, 1, 0, stream>>>(state, k);
    k_eqflag<<<cdiv(n, 256), 256, 0, stream>>>(keys, state, eq, n);
    scan_excl(eq, eqrank, bsum, n, stream);
    k_keep<<<cdiv(n, 256), 256, 0, stream>>>(keys, eqrank, state, keepf, n);
    scan_excl(keepf, pos, bsum, n, stream);
    float* xn = bufs[cur];
    k_compact<<<cdiv(n, 256), 256, 0, stream>>>(xcur, sc, keepf, pos, xn, newid, n, cout);
    k_remap<<<cdiv(NEDGES, 256), 256, 0, stream>>>(srcW, dstW, emW, newid, NEDGES);
    xcur = xn; cur ^= 1; n = k;
  }

  // Final conv (64 -> 17), BN result written straight into d_out.
  {
    float* y = bufs[cur];
    run_conv(xcur, y, (float*)d_out, xw, Wpad, deg, dinv, srcW, dstW, emW, accstats, ssbuf,
             convW[8], convB[8], normG[8], normB[8], n, CIN[8], COUT[8], stream);
  }
}